// QuantumLayer_5987184411289
// MI455X (gfx1250) — compile-verified
//
#include <hip/hip_runtime.h>

#define THREADS 256

// Branchless Cody-Waite cosine, ~1 ulp on |x| <~ 2^12 (inputs are N(0,1)).
// n = rint(x*2/pi); r = x - n*pi/2 (2-constant FMA reduction);
// cos(x) selected from Cephes sin/cos kernels on [-pi/4, pi/4] by quadrant.
__device__ __forceinline__ float fast_cos(float x) {
    const float TWO_OVER_PI = 0.636619772367581343f;
    const float PIO2_HI     = 1.57079637050628662109375f;   // 0x3FC90FDB
    const float PIO2_LO     = -4.37113882867379290e-08f;    // pi/2 - PIO2_HI

    float fn = __builtin_rintf(x * TWO_OVER_PI);            // v_rndne_f32
    int   k  = (int)fn;
    float r  = __builtin_fmaf(-fn, PIO2_HI, x);
    r        = __builtin_fmaf(-fn, PIO2_LO, r);
    float z  = r * r;

    // sin kernel: r + r^3 * (s0 + z*(s1 + z*s2))
    float sp = __builtin_fmaf(z, -1.9515295891e-4f, 8.3321608736e-3f);
    sp       = __builtin_fmaf(z, sp, -1.6666654611e-1f);
    sp       = __builtin_fmaf(z * r, sp, r);

    // cos kernel: 1 - z/2 + z^2*(c0 + z*(c1 + z*c2))
    float cp = __builtin_fmaf(z, 2.443315711809948e-5f, -1.388731625493765e-3f);
    cp       = __builtin_fmaf(z, cp, 4.166664568298827e-2f);
    cp       = __builtin_fmaf(z, cp, -0.5f);
    cp       = __builtin_fmaf(z, cp, 1.0f);

    float v = (k & 1) ? sp : cp;          // odd quadrant -> sin kernel
    return ((k + 1) & 2) ? -v : v;        // quadrants 1,2 -> negate
}

// out[:,0] = prod_{j=1..7} cos(theta_j)
// out[:,q] = prod_{j=0..q} cos(theta_j)   (q = 1..7)
// Exact closed form of the reference 8-qubit RY + CNOT-ring + <Z> circuit.
__global__ __launch_bounds__(THREADS) void qlayer_closed_form_kernel(
    const float* __restrict__ in,   // (B, 8) row-major angles
    float* __restrict__ out,        // (B, 8) row-major expectations
    int B) {
    __shared__ float4 tile[THREADS * 2];   // 32 B per thread = one sample

    const int gid = blockIdx.x * THREADS + threadIdx.x;
    if (gid >= B) return;

    // ---- gfx1250 async copy: global -> LDS (ASYNCcnt path) ----
    // INST_OFFSET is added to both the LDS and the memory address, so one
    // (lds_off, gaddr) pair + offset:16 stages both halves of the sample.
    const unsigned lds_off =
        (unsigned)(uintptr_t)(&tile[(unsigned)threadIdx.x * 2u]);
    const unsigned long long gaddr =
        (unsigned long long)(uintptr_t)(in + (size_t)gid * 8u);
    asm volatile(
        "global_load_async_to_lds_b128 %0, %1, off\n\t"
        "global_load_async_to_lds_b128 %0, %1, off offset:16\n\t"
        "s_wait_asynccnt 0"
        :
        : "v"(lds_off), "v"(gaddr)
        : "memory");

    const float4 t0 = tile[(unsigned)threadIdx.x * 2u + 0u];
    const float4 t1 = tile[(unsigned)threadIdx.x * 2u + 1u];

    const float c0 = fast_cos(t0.x);
    const float c1 = fast_cos(t0.y);
    const float c2 = fast_cos(t0.z);
    const float c3 = fast_cos(t0.w);
    const float c4 = fast_cos(t1.x);
    const float c5 = fast_cos(t1.y);
    const float c6 = fast_cos(t1.z);
    const float c7 = fast_cos(t1.w);

    // prefix products of cos(theta)
    const float p1 = c0 * c1;
    const float p2 = p1 * c2;
    const float p3 = p2 * c3;
    const float p4 = p3 * c4;
    const float p5 = p4 * c5;
    const float p6 = p5 * c6;
    const float p7 = p6 * c7;

    // out0 = c1*c2*...*c7 (no division: safe when c0 == 0)
    float r = c1 * c2;
    r *= c3;
    r *= c4;
    r *= c5;
    r *= c6;
    r *= c7;

    float4* dst = (float4*)out + (size_t)gid * 2u;
    dst[0] = make_float4(r, p1, p2, p3);
    dst[1] = make_float4(p4, p5, p6, p7);
}

extern "C" void kernel_launch(void* const* d_in, const int* in_sizes, int n_in,
                              void* d_out, int out_size, void* d_ws, size_t ws_size,
                              hipStream_t stream) {
    const float* in = (const float*)d_in[0];
    float* out = (float*)d_out;
    const int B = in_sizes[0] / 8;              // 524288 samples
    const int blocks = (B + THREADS - 1) / THREADS;
    qlayer_closed_form_kernel<<<blocks, THREADS, 0, stream>>>(in, out, B);
}